// Attention_Module_2559800508962
// MI455X (gfx1250) — compile-verified
//
#include <hip/hip_runtime.h>
#include <hip/hip_bf16.h>
#include <math.h>

// Problem constants (match reference).
#define BB     32
#define SS     2048
#define HH     1024
#define DD     1024
#define CHUNK  32
#define NCHUNK (SS / CHUNK)   // 64 chunks per batch

typedef float v2f __attribute__((ext_vector_type(2)));
typedef float v4f __attribute__((ext_vector_type(4)));
typedef float v8f __attribute__((ext_vector_type(8)));

// ---------------------------------------------------------------------------
// Kernel 1: x[32,1024] = hidden[32,1024] @ W1[1024,1024]^T  via WMMA f32 16x16x4
// grid = (DD/16, BB/16) = (64, 2), block = 32 (one wave per 16x16 tile)
// ---------------------------------------------------------------------------
__global__ __launch_bounds__(32)
void gemm1_wmma(const float* __restrict__ hidden,
                const float* __restrict__ W1,
                float* __restrict__ x)
{
    const int lane  = threadIdx.x;          // 0..31
    const int n0    = blockIdx.x * 16;      // output column tile (d)
    const int m0    = blockIdx.y * 16;      // output row tile (b)
    const int mrow  = m0 + (lane & 15);     // A: lane holds row M = lane%16
    const int nrow  = n0 + (lane & 15);     // B: lane holds col N = lane%16
    const int khalf = (lane >> 4) * 2;      // lanes 0-15: K+{0,1}; 16-31: K+{2,3}

    v8f c = {0.f, 0.f, 0.f, 0.f, 0.f, 0.f, 0.f, 0.f};

    for (int k = 0; k < HH; k += 4) {
        v2f a, bm;
        a.x  = hidden[mrow * HH + k + khalf + 0];
        a.y  = hidden[mrow * HH + k + khalf + 1];
        bm.x = W1[nrow * HH + k + khalf + 0];   // B[k][n] = W1[n][k]
        bm.y = W1[nrow * HH + k + khalf + 1];
        c = __builtin_amdgcn_wmma_f32_16x16x4_f32(
                /*neg_a=*/false, a, /*neg_b=*/false, bm,
                /*c_mod=*/(short)0, c, /*reuse_a=*/false, /*reuse_b=*/false);
    }

    // D layout: VGPR r -> M=r (lanes 0-15) / M=8+r (lanes 16-31), N = lane%16
    const int ncol  = n0 + (lane & 15);
    const int mbase = m0 + ((lane >> 4) << 3);
#pragma unroll
    for (int r = 0; r < 8; ++r)
        x[(mbase + r) * DD + ncol] = c[r];
}

// ---------------------------------------------------------------------------
// Streaming helpers for the attention pass.
// ---------------------------------------------------------------------------
__device__ __forceinline__ void load_row_nt(v4f (&r)[8], const float* row, int lane)
{
#pragma unroll
    for (int i = 0; i < 8; ++i)
        r[i] = __builtin_nontemporal_load((const v4f*)(row + lane * 4 + i * 128));
}

__device__ __forceinline__ void process_row(const v4f (&r)[8], const v4f (&xr)[8],
                                            v4f (&acc)[8], float& m, float& l,
                                            int s, int len, int lane,
                                            float* __restrict__ scorep)
{
    float dot = 0.0f;
#pragma unroll
    for (int i = 0; i < 8; ++i)
        dot += r[i].x * xr[i].x + r[i].y * xr[i].y
             + r[i].z * xr[i].z + r[i].w * xr[i].w;

    // wave32 butterfly reduction -> every lane holds the full dot product
#pragma unroll
    for (int off = 16; off > 0; off >>= 1)
        dot += __shfl_xor(dot, off, 32);

    float sc = (s < len) ? dot : 0.0f;
    if (sc == 0.0f) sc = -1e10f;            // reference quirk
    if (lane == 0) *scorep = sc;

    const float m_new = fmaxf(m, sc);
    const float scale = __expf(m - m_new);  // exp(-inf)=0 on first row
    const float w     = __expf(sc - m_new);
    l = l * scale + w;
#pragma unroll
    for (int i = 0; i < 8; ++i) {
        acc[i].x = fmaf(acc[i].x, scale, w * r[i].x);
        acc[i].y = fmaf(acc[i].y, scale, w * r[i].y);
        acc[i].z = fmaf(acc[i].z, scale, w * r[i].z);
        acc[i].w = fmaf(acc[i].w, scale, w * r[i].w);
    }
    m = m_new;
}

// ---------------------------------------------------------------------------
// Kernel 2: single streaming pass over encoder_outs (read ONCE, non-temporal).
// One wave per (batch b, chunk c of 32 source rows), double-buffered rows.
// grid = (NCHUNK, BB), block = 32.
// ---------------------------------------------------------------------------
__global__ __launch_bounds__(32)
void attn_pass(const float* __restrict__ enc,      // [B,S,D]
               const float* __restrict__ x,        // [B,D]
               const int*   __restrict__ src_lens, // [B]
               float* __restrict__ scores,         // [B,S] raw masked scores
               float* __restrict__ pm,             // [B,NCHUNK] chunk max
               float* __restrict__ pl,             // [B,NCHUNK] chunk expsum
               float* __restrict__ pctx)           // [B,NCHUNK,D] chunk ctx
{
    const int lane = threadIdx.x;
    const int c    = blockIdx.x;
    const int b    = blockIdx.y;
    const int s0   = c * CHUNK;
    const int len  = src_lens[b];

    // x_b resident in registers: element indices lane*4 + i*128 (coalesced f4)
    const float* xb = x + b * DD;
    v4f xr[8];
#pragma unroll
    for (int i = 0; i < 8; ++i)
        xr[i] = *(const v4f*)(xb + lane * 4 + i * 128);

    float m = -INFINITY, l = 0.0f;
    v4f acc[8];
#pragma unroll
    for (int i = 0; i < 8; ++i) acc[i] = (v4f){0.f, 0.f, 0.f, 0.f};

    const float* encb = enc + (size_t)b * SS * DD;

    // Double-buffered streaming: loads for row s+1 are in flight while row s
    // goes through the reduction/exp/accumulate dependency chain.
    v4f bufA[8], bufB[8];
    load_row_nt(bufA, encb + (size_t)s0 * DD, lane);

    for (int i = 0; i < CHUNK; i += 2) {
        const int s = s0 + i;
        load_row_nt(bufB, encb + (size_t)(s + 1) * DD, lane);
        process_row(bufA, xr, acc, m, l, s, len, lane, scores + b * SS + s);
        if (i + 2 < CHUNK)   // wave-uniform branch; avoids OOB read past chunk
            load_row_nt(bufA, encb + (size_t)(s + 2) * DD, lane);
        process_row(bufB, xr, acc, m, l, s + 1, len, lane, scores + b * SS + s + 1);
    }

    const int pidx = b * NCHUNK + c;
    if (lane == 0) { pm[pidx] = m; pl[pidx] = l; }
    float* pc = pctx + (size_t)pidx * DD;
#pragma unroll
    for (int i = 0; i < 8; ++i)
        *(v4f*)(pc + lane * 4 + i * 128) = acc[i];
}

// ---------------------------------------------------------------------------
// Kernel 3: per-batch combine of chunk partials -> final ctx + attn.T output.
// grid = BB, block = 256.
// ---------------------------------------------------------------------------
__global__ __launch_bounds__(256)
void combine(const float* __restrict__ pm,
             const float* __restrict__ pl,
             const float* __restrict__ pctx,
             const float* __restrict__ scores,
             float* __restrict__ ctx,     // [B,D]
             float* __restrict__ attnT)   // [S,B]
{
    __shared__ float sm[NCHUNK];
    __shared__ float sw[NCHUNK];
    __shared__ float sML[2];

    const int b = blockIdx.x;
    const int t = threadIdx.x;

    if (t < NCHUNK) { sm[t] = pm[b * NCHUNK + t]; sw[t] = pl[b * NCHUNK + t]; }
    __syncthreads();

    if (t == 0) {
        float M = -INFINITY;
        for (int c = 0; c < NCHUNK; ++c) M = fmaxf(M, sm[c]);
        float L = 0.0f;
        for (int c = 0; c < NCHUNK; ++c) L += sw[c] * __expf(sm[c] - M);
        sML[0] = M; sML[1] = L;
    }
    __syncthreads();

    const float M    = sML[0];
    const float Linv = 1.0f / sML[1];
    if (t < NCHUNK) sw[t] = __expf(sm[t] - M);   // per-chunk rescale factor
    __syncthreads();

    // final context
    for (int d = t; d < DD; d += 256) {
        float acc = 0.0f;
        for (int c = 0; c < NCHUNK; ++c)
            acc += pctx[(size_t)(b * NCHUNK + c) * DD + d] * sw[c];
        ctx[b * DD + d] = acc * Linv;
    }
    // attn.T output [S,B]
    for (int s = t; s < SS; s += 256)
        attnT[s * BB + b] = __expf(scores[b * SS + s] - M) * Linv;
}

// ---------------------------------------------------------------------------
// Kernel 4: out[32,1024] = tanh(concat(ctx,hidden)[32,2048] @ W2[1024,2048]^T)
// WMMA f32 16x16x4, K split over the two concat halves.
// grid = (DD/16, BB/16) = (64, 2), block = 32.
// ---------------------------------------------------------------------------
__global__ __launch_bounds__(32)
void gemm2_wmma(const float* __restrict__ ctx,
                const float* __restrict__ hidden,
                const float* __restrict__ W2,      // [1024, 2048]
                float* __restrict__ out)
{
    const int lane  = threadIdx.x;
    const int n0    = blockIdx.x * 16;
    const int m0    = blockIdx.y * 16;
    const int mrow  = m0 + (lane & 15);
    const int nrow  = n0 + (lane & 15);
    const int khalf = (lane >> 4) * 2;

    v8f c = {0.f, 0.f, 0.f, 0.f, 0.f, 0.f, 0.f, 0.f};

    // first half of K: A = ctx, B[k][n] = W2[n][k]
    for (int k = 0; k < HH; k += 4) {
        v2f a, bm;
        a.x  = ctx[mrow * DD + k + khalf + 0];
        a.y  = ctx[mrow * DD + k + khalf + 1];
        bm.x = W2[nrow * (HH + DD) + k + khalf + 0];
        bm.y = W2[nrow * (HH + DD) + k + khalf + 1];
        c = __builtin_amdgcn_wmma_f32_16x16x4_f32(false, a, false, bm,
                                                  (short)0, c, false, false);
    }
    // second half of K: A = hidden, B[k][n] = W2[n][1024 + k]
    for (int k = 0; k < HH; k += 4) {
        v2f a, bm;
        a.x  = hidden[mrow * HH + k + khalf + 0];
        a.y  = hidden[mrow * HH + k + khalf + 1];
        bm.x = W2[nrow * (HH + DD) + HH + k + khalf + 0];
        bm.y = W2[nrow * (HH + DD) + HH + k + khalf + 1];
        c = __builtin_amdgcn_wmma_f32_16x16x4_f32(false, a, false, bm,
                                                  (short)0, c, false, false);
    }

    const int ncol  = n0 + (lane & 15);
    const int mbase = m0 + ((lane >> 4) << 3);
#pragma unroll
    for (int r = 0; r < 8; ++r)
        out[(mbase + r) * DD + ncol] = tanhf(c[r]);
}

// ---------------------------------------------------------------------------
// Host-side launch. Inputs: hidden, encoder_outs, src_lens, W1, W2.
// Output: out[32,1024] ++ attn.T[2048,32], all fp32.
// ---------------------------------------------------------------------------
extern "C" void kernel_launch(void* const* d_in, const int* in_sizes, int n_in,
                              void* d_out, int out_size, void* d_ws, size_t ws_size,
                              hipStream_t stream) {
    const float* hidden   = (const float*)d_in[0];
    const float* enc      = (const float*)d_in[1];
    const int*   src_lens = (const int*)  d_in[2];
    const float* W1       = (const float*)d_in[3];
    const float* W2       = (const float*)d_in[4];

    float* out_ptr = (float*)d_out;                 // [32*1024]
    float* attnT   = out_ptr + BB * DD;             // [2048*32]

    // Workspace layout (floats): ~9 MB total
    float* ws        = (float*)d_ws;
    float* ws_x      = ws;                          // 32*1024
    float* ws_ctx    = ws_x    + BB * DD;           // 32*1024
    float* ws_scores = ws_ctx  + BB * DD;           // 32*2048
    float* ws_pm     = ws_scores + BB * SS;         // 32*64
    float* ws_pl     = ws_pm   + BB * NCHUNK;       // 32*64
    float* ws_pctx   = ws_pl   + BB * NCHUNK;       // 32*64*1024 (~8 MB)

    // 1) x = hidden @ W1^T
    gemm1_wmma<<<dim3(DD / 16, BB / 16), 32, 0, stream>>>(hidden, W1, ws_x);

    // 2) streaming scores + online-softmax partial contexts (enc read ONCE)
    attn_pass<<<dim3(NCHUNK, BB), 32, 0, stream>>>(enc, ws_x, src_lens,
                                                   ws_scores, ws_pm, ws_pl, ws_pctx);

    // 3) combine partials -> ctx, emit attn.T
    combine<<<dim3(BB), 256, 0, stream>>>(ws_pm, ws_pl, ws_pctx, ws_scores,
                                          ws_ctx, attnT);

    // 4) out = tanh(concat(ctx, hidden) @ W2^T)
    gemm2_wmma<<<dim3(DD / 16, BB / 16), 32, 0, stream>>>(ws_ctx, hidden, W2, out_ptr);
}